// GCNLayer_14637248544872
// MI455X (gfx1250) — compile-verified
//
#include <hip/hip_runtime.h>

typedef float v2f __attribute__((ext_vector_type(2)));
typedef float v8f __attribute__((ext_vector_type(8)));

#define C_OUT 32
#define C_IN  128

// ---------------- utility fills ----------------
__global__ void fill_f32_k(float* __restrict__ p, float v, long long n) {
    long long i = (long long)blockIdx.x * blockDim.x + threadIdx.x;
    long long stride = (long long)gridDim.x * blockDim.x;
    for (; i < n; i += stride) p[i] = v;
}

__global__ void fill_u32_k(unsigned* __restrict__ p, unsigned v, long long n) {
    long long i = (long long)blockIdx.x * blockDim.x + threadIdx.x;
    long long stride = (long long)gridDim.x * blockDim.x;
    for (; i < n; i += stride) p[i] = v;
}

// ---------------- degree / norm ----------------
__global__ void degree_k(const int* __restrict__ dst, unsigned* __restrict__ deg, long long E) {
    long long i = (long long)blockIdx.x * blockDim.x + threadIdx.x;
    long long stride = (long long)gridDim.x * blockDim.x;
    for (; i < E; i += stride) atomicAdd(&deg[dst[i]], 1u);
}

// deg (uint, incl. +1 self loop) -> dinv (float), in place
__global__ void dinv_k(unsigned* __restrict__ degu, float* __restrict__ dinv, int N) {
    int i = blockIdx.x * blockDim.x + threadIdx.x;
    int stride = gridDim.x * blockDim.x;
    for (; i < N; i += stride) {
        float d = (float)degu[i] + 1.0f;   // self loop
        dinv[i] = rsqrtf(d);               // d >= 1 always
    }
}

// ---------------- WMMA GEMM: H[N x 32] = X[N x K] @ W[K x 32] ----------------
// One wave -> 16x32 output tile (two 16x16 WMMA accumulators), block = 8 waves = 128 rows.
// V_WMMA_F32_16X16X4_F32: A 16x4, B 4x16, C 16x16 f32.
__global__ void __launch_bounds__(256)
gemm_wmma_k(const float* __restrict__ X, const float* __restrict__ W,
            float* __restrict__ H, int Nrows, int K) {
    // W staged in LDS, transposed: Wlds[col*K + k]  (max 32*128 f32 = 16 KB)
    __shared__ float Wlds[C_OUT * C_IN];
    for (int t = threadIdx.x; t < K * C_OUT; t += blockDim.x) {
        int r = t / C_OUT;         // k index
        int c = t - r * C_OUT;     // output col
        Wlds[c * K + r] = W[t];
    }
    __syncthreads();

    const int wave = threadIdx.x >> 5;
    const int lane = threadIdx.x & 31;
    const int half = lane >> 4;    // 0: lanes 0-15, 1: lanes 16-31
    const int l16  = lane & 15;

    const int tileM = (blockIdx.x * 8 + wave) * 16;

    // Clamp A-row so every wave runs the WMMA loop with EXEC all-ones.
    int rowA = tileM + l16;
    if (rowA >= Nrows) rowA = Nrows - 1;
    const float* __restrict__ xrow = X + (long long)rowA * K;

    v8f c0 = {0.f, 0.f, 0.f, 0.f, 0.f, 0.f, 0.f, 0.f};
    v8f c1 = {0.f, 0.f, 0.f, 0.f, 0.f, 0.f, 0.f, 0.f};

    for (int k = 0; k < K; k += 4) {
        // A frag: VGPR0 = K(k + 2*half), VGPR1 = K(k + 2*half + 1); lane%16 = M
        v2f a = *(const v2f*)(xrow + k + 2 * half);
        // B frags (cols 0-15 and 16-31); lane%16 = N, same K striping as A
        v2f blo = *(const v2f*)(&Wlds[l16 * K + k + 2 * half]);
        v2f bhi = *(const v2f*)(&Wlds[(16 + l16) * K + k + 2 * half]);
        c0 = __builtin_amdgcn_wmma_f32_16x16x4_f32(false, a, false, blo, (short)0, c0, false, false);
        c1 = __builtin_amdgcn_wmma_f32_16x16x4_f32(false, a, false, bhi, (short)0, c1, false, false);
    }

    // C/D layout: VGPR r -> row (r + 8*half), col = lane%16 (per n-tile)
    for (int r = 0; r < 8; ++r) {
        int row = tileM + r + 8 * half;
        if (row < Nrows) {
            float* __restrict__ hrow = H + (long long)row * C_OUT;
            hrow[l16]      = c0[r];
            hrow[16 + l16] = c1[r];
        }
    }
}

// ---------------- edge scatter: out[dst] += h[src] * dinv[src]*dinv[dst] ----------------
// One wave handles one edge's 32 channels: src/dst loads are wave-uniform,
// gather + atomic-add are contiguous 128B bursts.
__global__ void scatter_k(const float* __restrict__ h, const float* __restrict__ dinv,
                          const int* __restrict__ src, const int* __restrict__ dst,
                          float* out, long long E) {
    long long i = (long long)blockIdx.x * blockDim.x + threadIdx.x;
    long long stride = (long long)gridDim.x * blockDim.x;
    long long total = E * C_OUT;
    for (; i < total; i += stride) {
        long long e = i >> 5;
        int c = (int)(i & 31);
        int s = src[e];
        int d = dst[e];
        float nrm = dinv[s] * dinv[d];
        atomicAdd(out + (long long)d * C_OUT + c, h[(long long)s * C_OUT + c] * nrm);
    }
}

// ---------------- finalize: agg += h_self * dinv^2 + bias  (opt. relu), in place ----------------
__global__ void finalize_k(float* __restrict__ agg, const float* __restrict__ hself,
                           const float* __restrict__ dinv, const float* __restrict__ bias,
                           int N, int do_relu) {
    long long i = (long long)blockIdx.x * blockDim.x + threadIdx.x;
    long long stride = (long long)gridDim.x * blockDim.x;
    long long total = (long long)N * C_OUT;
    for (; i < total; i += stride) {
        int node = (int)(i >> 5);
        int c = (int)(i & 31);
        float dv = dinv[node];
        float v = agg[i] + hself[i] * (dv * dv) + bias[c];
        if (do_relu) v = fmaxf(v, 0.0f);
        agg[i] = v;
    }
}

extern "C" void kernel_launch(void* const* d_in, const int* in_sizes, int n_in,
                              void* d_out, int out_size, void* d_ws, size_t ws_size,
                              hipStream_t stream) {
    const float* x  = (const float*)d_in[0];
    const int*   ei = (const int*)d_in[1];   // [2, E] int32
    const float* W1 = (const float*)d_in[2];
    const float* b1 = (const float*)d_in[3];
    const float* W2 = (const float*)d_in[4];
    const float* b2 = (const float*)d_in[5];

    const int N = in_sizes[0] / C_IN;        // 100000
    const long long E = (long long)in_sizes[1] / 2;
    const int* src = ei;
    const int* dst = ei + E;

    // workspace layout
    char* ws = (char*)d_ws;
    size_t nodeF   = (size_t)N * sizeof(float);
    size_t featF   = (size_t)N * C_OUT * sizeof(float);
    size_t offDinv = 0;
    size_t offA    = (nodeF + 255) & ~(size_t)255;
    size_t offB    = offA + ((featF + 255) & ~(size_t)255);
    float* dinv = (float*)(ws + offDinv);
    float* bufA = (float*)(ws + offA);       // h1, later h2
    float* bufB = (float*)(ws + offB);       // agg1 -> act1
    float* outp = (float*)d_out;             // agg2 -> final

    const int T = 256;
    const int gemmBlocks = (N + 127) / 128;

    // degrees & symmetric norm
    fill_u32_k<<<(N + T - 1) / T, T, 0, stream>>>((unsigned*)dinv, 0u, N);
    degree_k<<<2048, T, 0, stream>>>(dst, (unsigned*)dinv, E);
    dinv_k<<<(N + T - 1) / T, T, 0, stream>>>((unsigned*)dinv, dinv, N);

    // ---- layer 1 ----
    gemm_wmma_k<<<gemmBlocks, 256, 0, stream>>>(x, W1, bufA, N, C_IN);
    fill_f32_k<<<2048, T, 0, stream>>>(bufB, 0.0f, (long long)N * C_OUT);
    scatter_k<<<4096, T, 0, stream>>>(bufA, dinv, src, dst, bufB, E);
    finalize_k<<<2048, T, 0, stream>>>(bufB, bufA, dinv, b1, N, 1);  // relu -> act1 in bufB

    // ---- layer 2 ----
    gemm_wmma_k<<<gemmBlocks, 256, 0, stream>>>(bufB, W2, bufA, N, C_OUT);
    fill_f32_k<<<2048, T, 0, stream>>>(outp, 0.0f, (long long)N * C_OUT);
    scatter_k<<<4096, T, 0, stream>>>(bufA, dinv, src, dst, outp, E);
    finalize_k<<<2048, T, 0, stream>>>(outp, bufA, dinv, b2, N, 0);
}